// MyPointNetConv_45097156608515
// MI455X (gfx1250) — compile-verified
//
#include <hip/hip_runtime.h>
#include <hip/hip_bf16.h>

typedef __attribute__((ext_vector_type(2))) float v2f;
typedef __attribute__((ext_vector_type(8))) float v8f;

#define ODIM 128
#define KDIM 64
#define XDIM 61
#define TILE_E 16

// Fold BN into weights and pre-swizzle W' into B-fragment layout:
// Bfrag[ntile][kstep][lane][j] = W'[ntile*16 + (lane&15)][kstep*4 + (lane>>4)*2 + j]
__global__ __launch_bounds__(256) void pnc_prep(
    const float* __restrict__ W, const float* __restrict__ gamma,
    const float* __restrict__ beta, const float* __restrict__ mean,
    const float* __restrict__ var, float* __restrict__ Bfrag,
    float* __restrict__ bias)
{
  int i = blockIdx.x * 256 + threadIdx.x;
  if (i < ODIM) {
    float s = gamma[i] * rsqrtf(var[i] + 1e-5f);
    bias[i] = beta[i] - mean[i] * s;
  }
  if (i < ODIM * KDIM) {                 // 8192 fragment entries
    int j     = i & 1;
    int lane  = (i >> 1) & 31;
    int kstep = (i >> 6) & 15;
    int nt    = i >> 10;
    int ch = nt * 16 + (lane & 15);
    int k  = kstep * 4 + ((lane >> 4) << 1) + j;
    float s = gamma[ch] * rsqrtf(var[ch] + 1e-5f);
    Bfrag[i] = W[ch * KDIM + k] * s;
  }
}

__global__ __launch_bounds__(256) void pnc_zero(float4* __restrict__ out, int n4) {
  int i = blockIdx.x * 256 + threadIdx.x;
  if (i < n4) out[i] = make_float4(0.f, 0.f, 0.f, 0.f);
}

// One block per 16-edge tile; 8 waves each own a 16-channel slice of the 128 outputs.
__global__ __launch_bounds__(256) void pnc_edge(
    const float* __restrict__ x, const float* __restrict__ pos,
    const int2* __restrict__ edges, const float* __restrict__ Bfrag,
    const float* __restrict__ bias, float* __restrict__ out, int E_)
{
  __shared__ float msg[TILE_E][68];   // stride 68 -> conflict-free A-fragment reads
  __shared__ int   tgtS[TILE_E];
  const int base = blockIdx.x * TILE_E;
  const int tid  = threadIdx.x;

  // Build 16x64 message tile in LDS (gather x[src], pos deltas)
  for (int i = tid; i < TILE_E * KDIM; i += 256) {
    int m = i >> 6;
    int k = i & 63;
    int ei = base + m; if (ei >= E_) ei = E_ - 1;   // duplicate edge: harmless under max
    int2 e = edges[ei];                              // e.x = tgt, e.y = src
    float v;
    if (k < XDIM) {
      v = x[(long)e.y * XDIM + k];
    } else {
      int d = k - XDIM;
      v = pos[e.y * 3 + d] - pos[e.x * 3 + d];
    }
    msg[m][k] = v;
  }
  if (tid < TILE_E) {
    int ei = base + tid; if (ei >= E_) ei = E_ - 1;
    tgtS[tid] = edges[ei].x;
  }
  __syncthreads();

  const int wave  = tid >> 5;          // channel tile 0..7
  const int lane  = tid & 31;
  const int m     = lane & 15;         // A-matrix row
  const int khalf = (lane >> 4) << 1;  // K sub-offset per half-wave
  v8f c = {};
  const float* bp = Bfrag + (wave * 16 * 32 + lane) * 2;

  #pragma unroll
  for (int ks = 0; ks < 16; ++ks) {    // K = 64 in steps of 4
    v2f a;
    a.x = msg[m][ks * 4 + khalf];
    a.y = msg[m][ks * 4 + khalf + 1];
    v2f b = *(const v2f*)(bp + ks * 64);   // coalesced 8B/lane from pre-swizzled W'
    c = __builtin_amdgcn_wmma_f32_16x16x4_f32(false, a, false, b,
                                              (short)0, c, false, false);
  }

  // Epilogue: D[m][n], m = r + 8*(lane>=16), n = lane&15; add bias, atomic max (ReLU via 0-init)
  const int n     = lane & 15;
  const int mbase = (lane >> 4) << 3;
  const int ch    = wave * 16 + n;
  const float bv  = bias[ch];
  #pragma unroll
  for (int r = 0; r < 8; ++r) {
    float v = c[r] + bv;
    float* addr = out + (long)tgtS[mbase + r] * ODIM + ch;
    asm volatile("global_atomic_max_num_f32 %0, %1, off scope:SCOPE_DEV"
                 :: "v"(addr), "v"(v) : "memory");
  }
}

extern "C" void kernel_launch(void* const* d_in, const int* in_sizes, int n_in,
                              void* d_out, int out_size, void* d_ws, size_t ws_size,
                              hipStream_t stream) {
  const float* x     = (const float*)d_in[0];
  const float* pos   = (const float*)d_in[1];
  const int2*  edges = (const int2*)d_in[2];
  const float* W     = (const float*)d_in[3];
  const float* gamma = (const float*)d_in[4];
  const float* beta  = (const float*)d_in[5];
  const float* mean  = (const float*)d_in[6];
  const float* var   = (const float*)d_in[7];
  float* out = (float*)d_out;

  float* Bfrag = (float*)d_ws;                                  // 8192 floats
  float* bias  = (float*)((char*)d_ws + 8192 * sizeof(float));  // 128 floats

  int E_ = in_sizes[2] / 2;
  int n4 = out_size / 4;   // ODIM*N divisible by 4

  pnc_prep<<<(ODIM * KDIM + 255) / 256, 256, 0, stream>>>(W, gamma, beta, mean, var, Bfrag, bias);
  pnc_zero<<<(n4 + 255) / 256, 256, 0, stream>>>((float4*)out, n4);

  int ntile = (E_ + TILE_E - 1) / TILE_E;
  pnc_edge<<<ntile, 256, 0, stream>>>(x, pos, edges, Bfrag, bias, out, E_);
}